// DGCNN_6700148981855
// MI455X (gfx1250) — compile-verified
//
#include <hip/hip_runtime.h>
#include <float.h>

// ---------------------------------------------------------------------------
// DGCNN on MI455X (gfx1250): WMMA f32 16x16x4 for every 1x1-conv GEMM,
// VALU kNN with LDS-staged neighbor chunks, BN fused via two-pass
// stats/affine, activations transposed through padded LDS for layer-2 GEMMs.
// All workgroups running WMMA are single-wave (32 threads) so EXEC is all-1s.
// Round-1 fixes: branch-free CIN=3 edge fragments (cndmask, no divergent
// loads), int4 neighbor-index batching (b128), kk-split stats over gridDim.z.
// ---------------------------------------------------------------------------

typedef __attribute__((ext_vector_type(2))) float v2f;
typedef __attribute__((ext_vector_type(8))) float v8f;

#define KNN_K 20
#define SLOPE 0.1f

__device__ __forceinline__ float lrelu_f(float a) { return fmaxf(a, SLOPE * a); }

// Sum across each 16-lane half (rows live per half in WMMA C/D layout).
__device__ __forceinline__ float half16_reduce(float v) {
  v += __shfl_xor(v, 1, 32);
  v += __shfl_xor(v, 2, 32);
  v += __shfl_xor(v, 4, 32);
  v += __shfl_xor(v, 8, 32);
  return v;
}

// A-fragment (16x4 f32) from LDS-resident weights, padded pitch (even).
// lane<16 -> K{0,1}, lane>=16 -> K{2,3}; row = 16*mt + (lane&15).
__device__ __forceinline__ v2f afrag_lds(const float* ldsW, int pitch, int mt,
                                         int ks, int lane) {
  int row = 16 * mt + (lane & 15);
  int q0  = 4 * ks + ((lane & 16) ? 2 : 0);
  const float2* p = (const float2*)(ldsW + row * pitch + q0);
  float2 t = *p;
  v2f a; a.x = t.x; a.y = t.y;
  return a;
}

// Edge feature element q of column (point rowi, neighbor rowj):
// [x_j - x_i (CIN) | x_i (CIN) | zero pad]
template <int CIN>
__device__ __forceinline__ float edge_elem(const float* __restrict__ x,
                                           int rowi, int rowj, int q) {
  if (q < CIN)     return x[rowj * CIN + q] - x[rowi * CIN + q];
  if (q < 2 * CIN) return x[rowi * CIN + (q - CIN)];
  return 0.0f;
}

template <int CIN>
__device__ __forceinline__ v2f edge_bfrag(const float* __restrict__ x,
                                          int rowi, int rowj, int ks, int lane) {
  int q0 = 4 * ks + ((lane & 16) ? 2 : 0);
  v2f b;
  b.x = edge_elem<CIN>(x, rowi, rowj, q0);
  b.y = edge_elem<CIN>(x, rowi, rowj, q0 + 1);
  return b;
}

// B-fragment from LDS activation tile h[col][ch], padded pitch (conflict-light).
__device__ __forceinline__ v2f h_bfrag(const float* ldsH, int pitch, int ks,
                                       int lane) {
  int col = lane & 15;
  int q0  = 4 * ks + ((lane & 16) ? 2 : 0);
  const float2* p = (const float2*)(ldsH + col * pitch + q0);
  float2 t = *p;
  v2f b; b.x = t.x; b.y = t.y;
  return b;
}

// Stage weight block [moff..moff+MB) x KREAL into LDS with zero-padded pitch.
template <int MB, int KREAL, int PK>
__device__ __forceinline__ void load_w_lds(float* ldsW,
                                           const float* __restrict__ W,
                                           int moff, int lane) {
  for (int e = lane; e < MB * PK; e += 32) {
    int r = e / PK, q = e - r * PK;
    ldsW[e] = (q < KREAL) ? W[(moff + r) * KREAL + q] : 0.0f;
  }
}

// GEMM: acc[MT] (out ch x 16 cols) += W * edge(x, idx).
// CIN==3 path is branch-free: unconditional 3+3 float loads, cndmask selects.
template <int CIN, int MT, int KS, int PK>
__device__ __forceinline__ void gemm_edge(v8f (&acc)[MT],
                                          const float* __restrict__ x,
                                          int rowi, int rowj,
                                          const float* ldsW, int lane) {
  if constexpr (CIN == 3) {
    float xi0 = x[rowi * 3 + 0], xi1 = x[rowi * 3 + 1], xi2 = x[rowi * 3 + 2];
    float xj0 = x[rowj * 3 + 0], xj1 = x[rowj * 3 + 1], xj2 = x[rowj * 3 + 2];
    float d0 = xj0 - xi0, d1 = xj1 - xi1, d2 = xj2 - xi2;
    bool hi = (lane & 16) != 0;
    // edge vec e[8] = {d0,d1,d2, xi0,xi1,xi2, 0,0}
    v2f b0, b1;
    b0.x = hi ? d2 : d0;   b0.y = hi ? xi0 : d1;   // ks=0: q0 = hi?2:0
    b1.x = hi ? 0.f : xi1; b1.y = hi ? 0.f : xi2;  // ks=1: q0 = hi?6:4
#pragma unroll
    for (int mt = 0; mt < MT; ++mt) {
      v2f af = afrag_lds(ldsW, PK, mt, 0, lane);
      acc[mt] = __builtin_amdgcn_wmma_f32_16x16x4_f32(
          false, af, false, b0, (short)0, acc[mt], false, false);
    }
#pragma unroll
    for (int mt = 0; mt < MT; ++mt) {
      v2f af = afrag_lds(ldsW, PK, mt, 1, lane);
      acc[mt] = __builtin_amdgcn_wmma_f32_16x16x4_f32(
          false, af, false, b1, (short)0, acc[mt], false, false);
    }
  } else {
#pragma unroll
    for (int ks = 0; ks < KS; ++ks) {
      v2f bf = edge_bfrag<CIN>(x, rowi, rowj, ks, lane);
#pragma unroll
      for (int mt = 0; mt < MT; ++mt) {
        v2f af = afrag_lds(ldsW, PK, mt, ks, lane);
        acc[mt] = __builtin_amdgcn_wmma_f32_16x16x4_f32(
            false, af, false, bf, (short)0, acc[mt], false, false);
      }
    }
  }
}

// GEMM: acc[MT] += W * h  (h = activation tile in LDS, K = M1)
template <int MT, int KS, int PK, int PH>
__device__ __forceinline__ void gemm_h(v8f (&acc)[MT], const float* ldsH,
                                       const float* ldsW, int lane) {
#pragma unroll
  for (int ks = 0; ks < KS; ++ks) {
    v2f bf = h_bfrag(ldsH, PH, ks, lane);
#pragma unroll
    for (int mt = 0; mt < MT; ++mt) {
      v2f af = afrag_lds(ldsW, PK, mt, ks, lane);
      acc[mt] = __builtin_amdgcn_wmma_f32_16x16x4_f32(
          false, af, false, bf, (short)0, acc[mt], false, false);
    }
  }
}

__device__ __forceinline__ int i4_elem(int4 v, int u) {
  int a[4] = {v.x, v.y, v.z, v.w};
  return a[u];  // only used with unrolled constant u -> stays in registers
}

// ---------------------------------------------------------------------------
// kNN: one lane per query row, neighbor chunk staged in LDS (broadcast reads),
// register-resident unrolled insertion into sorted top-20 (smallest sq dist).
// ---------------------------------------------------------------------------
template <int CIN>
__global__ __launch_bounds__(256) void knn_kernel(const float* __restrict__ x,
                                                  int* __restrict__ idxo,
                                                  int N) {
  constexpr int CH = 128;
  __shared__ float lx[CH * CIN];
  int bpb  = N / 256;
  int b    = blockIdx.x / bpb;
  int i    = (blockIdx.x % bpb) * 256 + threadIdx.x;
  int rowi = b * N + i;

  float xi[CIN];
#pragma unroll
  for (int c = 0; c < CIN; ++c) xi[c] = x[rowi * CIN + c];

  float dv[KNN_K];
  int   di[KNN_K];
#pragma unroll
  for (int t = 0; t < KNN_K; ++t) { dv[t] = FLT_MAX; di[t] = 0; }

  for (int j0 = 0; j0 < N; j0 += CH) {
    __syncthreads();
    for (int e = threadIdx.x; e < CH * CIN; e += 256)
      lx[e] = x[(b * N + j0) * CIN + e];
    __syncthreads();
    for (int jj = 0; jj < CH; ++jj) {
      float d = 0.0f;
#pragma unroll
      for (int c = 0; c < CIN; ++c) {
        float t = xi[c] - lx[jj * CIN + c];
        d = fmaf(t, t, d);
      }
      if (d < dv[KNN_K - 1]) {
        float cv = d; int ci = j0 + jj;
#pragma unroll
        for (int t = 0; t < KNN_K; ++t) {
          bool sw = cv < dv[t];
          float tv = dv[t]; int ti = di[t];
          dv[t] = sw ? cv : tv; di[t] = sw ? ci : ti;
          cv    = sw ? tv : cv; ci    = sw ? ti : ci;
        }
      }
    }
  }
#pragma unroll
  for (int t = 0; t < KNN_K; ++t) idxo[rowi * KNN_K + t] = di[t];
}

// ---------------------------------------------------------------------------
// Layer stats: y = W * edge(x); accumulate per-channel sum / sumsq (atomics).
// kk groups of 4 split across gridDim.z (atomics merge partials).
// ---------------------------------------------------------------------------
template <int CIN, int MB>
__global__ __launch_bounds__(32) void edge_stats_kernel(
    const float* __restrict__ x, const int* __restrict__ idx,
    const float* __restrict__ W, float* __restrict__ sum,
    float* __restrict__ sq, int N) {
  constexpr int KE = ((2 * CIN + 3) & ~3);
  constexpr int PK = KE + 4;
  constexpr int MT = MB / 16;
  constexpr int KS = KE / 4;
  __shared__ float ldsW[MB * PK];
  int lane = threadIdx.x;
  int moff = blockIdx.y * MB;
  load_w_lds<MB, 2 * CIN, PK>(ldsW, W, moff, lane);
  __syncthreads();

  int tpb  = N / 16;
  int b    = blockIdx.x / tpb;
  int p0   = (blockIdx.x % tpb) * 16;
  int col  = lane & 15;
  int rowi = b * N + p0 + col;
  int half8 = (lane & 16) ? 8 : 0;
  const int4* idx4 = (const int4*)idx;  // 20 ints/row, 16B aligned

  v8f s[MT] = {}; v8f s2[MT] = {};
  for (int kk4 = blockIdx.z; kk4 < KNN_K / 4; kk4 += gridDim.z) {
    int4 j4 = idx4[rowi * (KNN_K / 4) + kk4];
#pragma unroll
    for (int u = 0; u < 4; ++u) {
      int rowj = b * N + i4_elem(j4, u);
      v8f acc[MT] = {};
      gemm_edge<CIN, MT, KS, PK>(acc, x, rowi, rowj, ldsW, lane);
#pragma unroll
      for (int mt = 0; mt < MT; ++mt) {
        s[mt]  += acc[mt];
        s2[mt] += acc[mt] * acc[mt];
      }
    }
  }
#pragma unroll
  for (int mt = 0; mt < MT; ++mt)
#pragma unroll
    for (int v = 0; v < 8; ++v) {
      float a = half16_reduce(s[mt][v]);
      float q = half16_reduce(s2[mt][v]);
      if ((lane & 15) == 0) {
        int ch = moff + 16 * mt + v + half8;
        atomicAdd(&sum[ch], a);
        atomicAdd(&sq[ch], q);
      }
    }
}

// ---------------------------------------------------------------------------
// Two-layer stats: h1 = lrelu(affine1(W1*edge)); accumulate stats of W2*h1.
// h1 transposed through padded LDS tile for second GEMM's B operand.
// ---------------------------------------------------------------------------
template <int CIN, int M1, int M2>
__global__ __launch_bounds__(32) void edge_stats2_kernel(
    const float* __restrict__ x, const int* __restrict__ idx,
    const float* __restrict__ W1, const float* __restrict__ W2,
    const float* __restrict__ sc1, const float* __restrict__ sh1,
    float* __restrict__ sum, float* __restrict__ sq, int N) {
  constexpr int KE  = ((2 * CIN + 3) & ~3);
  constexpr int PK1 = KE + 4, PK2 = M1 + 4, PH = M1 + 4;
  constexpr int MT1 = M1 / 16, MT2 = M2 / 16;
  constexpr int KS1 = KE / 4, KS2 = M1 / 4;
  __shared__ float ldsW1[M1 * PK1];
  __shared__ float ldsW2[M2 * PK2];
  __shared__ float ldsH[16 * PH];
  __shared__ float ldsS1[M1], ldsT1[M1];
  int lane = threadIdx.x;
  load_w_lds<M1, 2 * CIN, PK1>(ldsW1, W1, 0, lane);
  load_w_lds<M2, M1, PK2>(ldsW2, W2, 0, lane);
  for (int e = lane; e < M1; e += 32) { ldsS1[e] = sc1[e]; ldsT1[e] = sh1[e]; }
  __syncthreads();

  int tpb  = N / 16;
  int b    = blockIdx.x / tpb;
  int p0   = (blockIdx.x % tpb) * 16;
  int col  = lane & 15;
  int rowi = b * N + p0 + col;
  int half8 = (lane & 16) ? 8 : 0;
  const int4* idx4 = (const int4*)idx;

  v8f s[MT2] = {}; v8f s2[MT2] = {};
  for (int kk4 = blockIdx.z; kk4 < KNN_K / 4; kk4 += gridDim.z) {
    int4 j4 = idx4[rowi * (KNN_K / 4) + kk4];
#pragma unroll
    for (int u = 0; u < 4; ++u) {
      int rowj = b * N + i4_elem(j4, u);
      v8f a1[MT1] = {};
      gemm_edge<CIN, MT1, KS1, PK1>(a1, x, rowi, rowj, ldsW1, lane);
#pragma unroll
      for (int mt = 0; mt < MT1; ++mt)
#pragma unroll
        for (int v = 0; v < 8; ++v) {
          int ch = 16 * mt + v + half8;
          float t = fmaf(a1[mt][v], ldsS1[ch], ldsT1[ch]);
          ldsH[col * PH + ch] = lrelu_f(t);
        }
      __syncthreads();
      v8f a2[MT2] = {};
      gemm_h<MT2, KS2, PK2, PH>(a2, ldsH, ldsW2, lane);
      __syncthreads();
#pragma unroll
      for (int mt = 0; mt < MT2; ++mt) {
        s[mt]  += a2[mt];
        s2[mt] += a2[mt] * a2[mt];
      }
    }
  }
#pragma unroll
  for (int mt = 0; mt < MT2; ++mt)
#pragma unroll
    for (int v = 0; v < 8; ++v) {
      float a = half16_reduce(s[mt][v]);
      float q = half16_reduce(s2[mt][v]);
      if ((lane & 15) == 0) {
        int ch = 16 * mt + v + half8;
        atomicAdd(&sum[ch], a);
        atomicAdd(&sq[ch], q);
      }
    }
}

// ---------------------------------------------------------------------------
// Two-layer final: recompute chain, apply both affines + lrelu, max over k.
// ---------------------------------------------------------------------------
template <int CIN, int M1, int M2>
__global__ __launch_bounds__(32) void edge_final2_kernel(
    const float* __restrict__ x, const int* __restrict__ idx,
    const float* __restrict__ W1, const float* __restrict__ W2,
    const float* __restrict__ sc1, const float* __restrict__ sh1,
    const float* __restrict__ sc2, const float* __restrict__ sh2,
    float* __restrict__ xout, int N) {
  constexpr int KE  = ((2 * CIN + 3) & ~3);
  constexpr int PK1 = KE + 4, PK2 = M1 + 4, PH = M1 + 4;
  constexpr int MT1 = M1 / 16, MT2 = M2 / 16;
  constexpr int KS1 = KE / 4, KS2 = M1 / 4;
  __shared__ float ldsW1[M1 * PK1];
  __shared__ float ldsW2[M2 * PK2];
  __shared__ float ldsH[16 * PH];
  __shared__ float ldsS1[M1], ldsT1[M1], ldsS2[M2], ldsT2[M2];
  int lane = threadIdx.x;
  load_w_lds<M1, 2 * CIN, PK1>(ldsW1, W1, 0, lane);
  load_w_lds<M2, M1, PK2>(ldsW2, W2, 0, lane);
  for (int e = lane; e < M1; e += 32) { ldsS1[e] = sc1[e]; ldsT1[e] = sh1[e]; }
  for (int e = lane; e < M2; e += 32) { ldsS2[e] = sc2[e]; ldsT2[e] = sh2[e]; }
  __syncthreads();

  int tpb  = N / 16;
  int b    = blockIdx.x / tpb;
  int p0   = (blockIdx.x % tpb) * 16;
  int col  = lane & 15;
  int rowi = b * N + p0 + col;
  int half8 = (lane & 16) ? 8 : 0;
  const int4* idx4 = (const int4*)idx;

  v8f mx[MT2];
#pragma unroll
  for (int mt = 0; mt < MT2; ++mt)
#pragma unroll
    for (int v = 0; v < 8; ++v) mx[mt][v] = -FLT_MAX;

  for (int kk4 = 0; kk4 < KNN_K / 4; ++kk4) {
    int4 j4 = idx4[rowi * (KNN_K / 4) + kk4];
#pragma unroll
    for (int u = 0; u < 4; ++u) {
      int rowj = b * N + i4_elem(j4, u);
      v8f a1[MT1] = {};
      gemm_edge<CIN, MT1, KS1, PK1>(a1, x, rowi, rowj, ldsW1, lane);
#pragma unroll
      for (int mt = 0; mt < MT1; ++mt)
#pragma unroll
        for (int v = 0; v < 8; ++v) {
          int ch = 16 * mt + v + half8;
          float t = fmaf(a1[mt][v], ldsS1[ch], ldsT1[ch]);
          ldsH[col * PH + ch] = lrelu_f(t);
        }
      __syncthreads();
      v8f a2[MT2] = {};
      gemm_h<MT2, KS2, PK2, PH>(a2, ldsH, ldsW2, lane);
      __syncthreads();
#pragma unroll
      for (int mt = 0; mt < MT2; ++mt)
#pragma unroll
        for (int v = 0; v < 8; ++v) {
          int ch = 16 * mt + v + half8;
          float t = fmaf(a2[mt][v], ldsS2[ch], ldsT2[ch]);
          mx[mt][v] = fmaxf(mx[mt][v], lrelu_f(t));
        }
    }
  }
#pragma unroll
  for (int mt = 0; mt < MT2; ++mt)
#pragma unroll
    for (int v = 0; v < 8; ++v) {
      int ch = 16 * mt + v + half8;
      xout[rowi * M2 + ch] = mx[mt][v];
    }
}

// ---------------------------------------------------------------------------
// Single-layer final (layer 5, M split over blockIdx.y): affine+lrelu+max.
// ---------------------------------------------------------------------------
template <int CIN, int MB>
__global__ __launch_bounds__(32) void edge_final1_kernel(
    const float* __restrict__ x, const int* __restrict__ idx,
    const float* __restrict__ W, const float* __restrict__ sc,
    const float* __restrict__ sh, float* __restrict__ xout, int outC, int N) {
  constexpr int KE = ((2 * CIN + 3) & ~3);
  constexpr int PK = KE + 4;
  constexpr int MT = MB / 16;
  constexpr int KS = KE / 4;
  __shared__ float ldsW[MB * PK];
  __shared__ float ldsS[MB], ldsT[MB];
  int lane = threadIdx.x;
  int moff = blockIdx.y * MB;
  load_w_lds<MB, 2 * CIN, PK>(ldsW, W, moff, lane);
  for (int e = lane; e < MB; e += 32) { ldsS[e] = sc[moff + e]; ldsT[e] = sh[moff + e]; }
  __syncthreads();

  int tpb  = N / 16;
  int b    = blockIdx.x / tpb;
  int p0   = (blockIdx.x % tpb) * 16;
  int col  = lane & 15;
  int rowi = b * N + p0 + col;
  int half8 = (lane & 16) ? 8 : 0;
  const int4* idx4 = (const int4*)idx;

  v8f mx[MT];
#pragma unroll
  for (int mt = 0; mt < MT; ++mt)
#pragma unroll
    for (int v = 0; v < 8; ++v) mx[mt][v] = -FLT_MAX;

  for (int kk4 = 0; kk4 < KNN_K / 4; ++kk4) {
    int4 j4 = idx4[rowi * (KNN_K / 4) + kk4];
#pragma unroll
    for (int u = 0; u < 4; ++u) {
      int rowj = b * N + i4_elem(j4, u);
      v8f acc[MT] = {};
      gemm_edge<CIN, MT, KS, PK>(acc, x, rowi, rowj, ldsW, lane);
#pragma unroll
      for (int mt = 0; mt < MT; ++mt)
#pragma unroll
        for (int v = 0; v < 8; ++v) {
          int ch = 16 * mt + v + half8;
          float t = fmaf(acc[mt][v], ldsS[ch], ldsT[ch]);
          mx[mt][v] = fmaxf(mx[mt][v], lrelu_f(t));
        }
    }
  }
#pragma unroll
  for (int mt = 0; mt < MT; ++mt)
#pragma unroll
    for (int v = 0; v < 8; ++v) {
      int ch = 16 * mt + v + half8;
      xout[rowi * outC + moff + ch] = mx[mt][v];
    }
}

// ---------------------------------------------------------------------------
// Head (layer 6): columns = points, K = 256 from concat(x1,x2,x3).
// ---------------------------------------------------------------------------
__device__ __forceinline__ float cat_elem(const float* __restrict__ x1,
                                          const float* __restrict__ x2,
                                          const float* __restrict__ x3,
                                          int row, int q) {
  if (q < 64)  return x1[row * 64 + q];
  if (q < 128) return x2[row * 64 + (q - 64)];
  return x3[row * 128 + (q - 128)];
}

__device__ __forceinline__ v2f cat_bfrag(const float* __restrict__ x1,
                                         const float* __restrict__ x2,
                                         const float* __restrict__ x3,
                                         int row, int ks, int lane) {
  int q0 = 4 * ks + ((lane & 16) ? 2 : 0);
  v2f b;
  b.x = cat_elem(x1, x2, x3, row, q0);
  b.y = cat_elem(x1, x2, x3, row, q0 + 1);
  return b;
}

template <int MB>
__global__ __launch_bounds__(32) void concat_stats_kernel(
    const float* __restrict__ x1, const float* __restrict__ x2,
    const float* __restrict__ x3, const float* __restrict__ W,
    float* __restrict__ sum, float* __restrict__ sq, int P) {
  constexpr int K = 256, PK = K + 4, MT = MB / 16, KS = K / 4;
  __shared__ float ldsW[MB * PK];
  int lane = threadIdx.x;
  int moff = blockIdx.y * MB;
  load_w_lds<MB, K, PK>(ldsW, W, moff, lane);
  __syncthreads();
  int col = lane & 15, half8 = (lane & 16) ? 8 : 0;

  v8f s[MT] = {}; v8f s2[MT] = {};
  int ntiles = P / 16;
  for (int t = blockIdx.x; t < ntiles; t += gridDim.x) {
    int row = t * 16 + col;
    v8f acc[MT] = {};
#pragma unroll
    for (int ks = 0; ks < KS; ++ks) {
      v2f bf = cat_bfrag(x1, x2, x3, row, ks, lane);
#pragma unroll
      for (int mt = 0; mt < MT; ++mt) {
        v2f af = afrag_lds(ldsW, PK, mt, ks, lane);
        acc[mt] = __builtin_amdgcn_wmma_f32_16x16x4_f32(
            false, af, false, bf, (short)0, acc[mt], false, false);
      }
    }
#pragma unroll
    for (int mt = 0; mt < MT; ++mt) {
      s[mt]  += acc[mt];
      s2[mt] += acc[mt] * acc[mt];
    }
  }
#pragma unroll
  for (int mt = 0; mt < MT; ++mt)
#pragma unroll
    for (int v = 0; v < 8; ++v) {
      float a = half16_reduce(s[mt][v]);
      float q = half16_reduce(s2[mt][v]);
      if ((lane & 15) == 0) {
        int ch = moff + 16 * mt + v + half8;
        atomicAdd(&sum[ch], a);
        atomicAdd(&sq[ch], q);
      }
    }
}

template <int MB>
__global__ __launch_bounds__(32) void concat_final_kernel(
    const float* __restrict__ x1, const float* __restrict__ x2,
    const float* __restrict__ x3, const float* __restrict__ W,
    const float* __restrict__ sc, const float* __restrict__ sh,
    float* __restrict__ out, int N, int P) {
  constexpr int K = 256, PK = K + 4, MT = MB / 16, KS = K / 4;
  __shared__ float ldsW[MB * PK];
  __shared__ float ldsS[MB], ldsT[MB];
  int lane = threadIdx.x;
  int moff = blockIdx.y * MB;
  load_w_lds<MB, K, PK>(ldsW, W, moff, lane);
  for (int e = lane; e < MB; e += 32) { ldsS[e] = sc[moff + e]; ldsT[e] = sh[moff + e]; }
  __syncthreads();
  int col = lane & 15, half8 = (lane & 16) ? 8 : 0;

  int row = blockIdx.x * 16 + col;
  int b = row / N, p = row - b * N;
  v8f acc[MT] = {};
#pragma unroll
  for (int ks = 0; ks < KS; ++ks) {
    v2f bf = cat_bfrag(x1, x2, x3, row, ks, lane);
#pragma unroll
    for (int mt = 0; mt < MT; ++mt) {
      v2f af = afrag_lds(ldsW, PK, mt, ks, lane);
      acc[mt] = __builtin_amdgcn_wmma_f32_16x16x4_f32(
          false, af, false, bf, (short)0, acc[mt], false, false);
    }
  }
#pragma unroll
  for (int mt = 0; mt < MT; ++mt)
#pragma unroll
    for (int v = 0; v < 8; ++v) {
      int chl = 16 * mt + v + half8;
      int chg = moff + chl;
      float t = fmaf(acc[mt][v], ldsS[chl], ldsT[chl]);
      out[((size_t)b * 512 + chg) * (size_t)N + p] = lrelu_f(t);
    }
}

// ---------------------------------------------------------------------------
// BN affine fold: scale = g * rsqrt(var+eps), shift = b - mean*scale.
// ---------------------------------------------------------------------------
__global__ void affine_kernel(const float* __restrict__ sum,
                              const float* __restrict__ sq,
                              const float* __restrict__ g,
                              const float* __restrict__ b,
                              float* __restrict__ sc, float* __restrict__ sh,
                              int M, float invCnt) {
  int c = blockIdx.x * blockDim.x + threadIdx.x;
  if (c < M) {
    float m = sum[c] * invCnt;
    float v = fmaf(sq[c], invCnt, -m * m);
    float s = g[c] * rsqrtf(v + 1e-5f);
    sc[c] = s;
    sh[c] = fmaf(-m, s, b[c]);
  }
}

// ---------------------------------------------------------------------------
extern "C" void kernel_launch(void* const* d_in, const int* in_sizes, int n_in,
                              void* d_out, int out_size, void* d_ws,
                              size_t ws_size, hipStream_t stream) {
  (void)in_sizes; (void)n_in; (void)out_size; (void)ws_size;
  const int B = 8, N = 2048;
  const int P = B * N;

  const float* x  = (const float*)d_in[0];
  const float* w1 = (const float*)d_in[1];
  const float* w2 = (const float*)d_in[2];
  const float* w3 = (const float*)d_in[3];
  const float* w4 = (const float*)d_in[4];
  const float* w5 = (const float*)d_in[5];
  const float* w6 = (const float*)d_in[6];
  const float* g[6]  = {(const float*)d_in[7],  (const float*)d_in[9],
                        (const float*)d_in[11], (const float*)d_in[13],
                        (const float*)d_in[15], (const float*)d_in[17]};
  const float* bb[6] = {(const float*)d_in[8],  (const float*)d_in[10],
                        (const float*)d_in[12], (const float*)d_in[14],
                        (const float*)d_in[16], (const float*)d_in[18]};
  float* out = (float*)d_out;

  char* ws = (char*)d_ws;
  size_t off = 0;
  int* idxb = (int*)(ws + off);      off += (size_t)P * KNN_K * sizeof(int);
  float* x1 = (float*)(ws + off);    off += (size_t)P * 64  * sizeof(float);
  float* x2 = (float*)(ws + off);    off += (size_t)P * 64  * sizeof(float);
  float* x3 = (float*)(ws + off);    off += (size_t)P * 128 * sizeof(float);
  float* st = (float*)(ws + off);
  float *sum1 = st + 0,    *sq1 = st + 64;
  float *sum2 = st + 128,  *sq2 = st + 192;
  float *sum3 = st + 256,  *sq3 = st + 320;
  float *sum4 = st + 384,  *sq4 = st + 448;
  float *sum5 = st + 512,  *sq5 = st + 640;
  float *sum6 = st + 768,  *sq6 = st + 1280;
  float *sc1 = st + 1792, *sh1 = st + 1856;
  float *sc2 = st + 1920, *sh2 = st + 1984;
  float *sc3 = st + 2048, *sh3 = st + 2112;
  float *sc4 = st + 2176, *sh4 = st + 2240;
  float *sc5 = st + 2304, *sh5 = st + 2432;
  float *sc6 = st + 2560, *sh6 = st + 3072;

  hipMemsetAsync(st, 0, 1792 * sizeof(float), stream);

  dim3 w32(32, 1, 1);
  int tilesX = P / 16;  // 1024
  float invE = 1.0f / ((float)P * (float)KNN_K);
  float invP = 1.0f / (float)P;

  // ---- stage 1: x(3) -> x1(64) ----
  knn_kernel<3><<<P / 256, 256, 0, stream>>>(x, idxb, N);
  edge_stats_kernel<3, 64><<<dim3(tilesX, 1, 5), w32, 0, stream>>>(x, idxb, w1, sum1, sq1, N);
  affine_kernel<<<1, 64, 0, stream>>>(sum1, sq1, g[0], bb[0], sc1, sh1, 64, invE);
  edge_stats2_kernel<3, 64, 64><<<dim3(tilesX, 1, 5), w32, 0, stream>>>(x, idxb, w1, w2, sc1, sh1, sum2, sq2, N);
  affine_kernel<<<1, 64, 0, stream>>>(sum2, sq2, g[1], bb[1], sc2, sh2, 64, invE);
  edge_final2_kernel<3, 64, 64><<<dim3(tilesX, 1), w32, 0, stream>>>(x, idxb, w1, w2, sc1, sh1, sc2, sh2, x1, N);

  // ---- stage 2: x1(64) -> x2(64) ----
  knn_kernel<64><<<P / 256, 256, 0, stream>>>(x1, idxb, N);
  edge_stats_kernel<64, 64><<<dim3(tilesX, 1, 5), w32, 0, stream>>>(x1, idxb, w3, sum3, sq3, N);
  affine_kernel<<<1, 64, 0, stream>>>(sum3, sq3, g[2], bb[2], sc3, sh3, 64, invE);
  edge_stats2_kernel<64, 64, 64><<<dim3(tilesX, 1, 5), w32, 0, stream>>>(x1, idxb, w3, w4, sc3, sh3, sum4, sq4, N);
  affine_kernel<<<1, 64, 0, stream>>>(sum4, sq4, g[3], bb[3], sc4, sh4, 64, invE);
  edge_final2_kernel<64, 64, 64><<<dim3(tilesX, 1), w32, 0, stream>>>(x1, idxb, w3, w4, sc3, sh3, sc4, sh4, x2, N);

  // ---- stage 3: x2(64) -> x3(128), single conv ----
  knn_kernel<64><<<P / 256, 256, 0, stream>>>(x2, idxb, N);
  edge_stats_kernel<64, 64><<<dim3(tilesX, 2, 5), w32, 0, stream>>>(x2, idxb, w5, sum5, sq5, N);
  affine_kernel<<<2, 64, 0, stream>>>(sum5, sq5, g[4], bb[4], sc5, sh5, 128, invE);
  edge_final1_kernel<64, 64><<<dim3(tilesX, 2), w32, 0, stream>>>(x2, idxb, w5, sc5, sh5, x3, 128, N);

  // ---- head: concat(256) -> out(512) ----
  concat_stats_kernel<32><<<dim3(64, 16), w32, 0, stream>>>(x1, x2, x3, w6, sum6, sq6, P);
  affine_kernel<<<8, 64, 0, stream>>>(sum6, sq6, g[5], bb[5], sc6, sh6, 512, invP);
  concat_final_kernel<32><<<dim3(tilesX, 16), w32, 0, stream>>>(x1, x2, x3, w6, sc6, sh6, out, N, P);
}